// CellConditionedCrossContextAttn_86122684220177
// MI455X (gfx1250) — compile-verified
//
#include <hip/hip_runtime.h>
#include <hip/hip_bf16.h>

// ---------------------------------------------------------------------------
// CellConditionedCrossContextAttn for MI455X (gfx1250, wave32, WMMA f32)
// B=32, LA=LB=128, D=256, LC=16, CIN=512, H=8, DH=32
// ---------------------------------------------------------------------------

typedef float v2f __attribute__((ext_vector_type(2)));
typedef float v8f __attribute__((ext_vector_type(8)));

#define Bn    32
#define Ln    128
#define Dn    256
#define LCn   16
#define CINn  512
#define Hn    8
#define DHn   32
#define HALF_ELEMS (Bn * Ln * Dn)          // 1,048,576 floats per (a|b) half
#define EPSv  1e-5f
#define NEGv  -1e9f
#define QSCALE 0.17677669529663687f        // 1/sqrt(32)

// ---------------- block-wide sum over 256 threads (8 waves) ----------------
__device__ __forceinline__ float block_sum_256(float v, float* red) {
#pragma unroll
  for (int o = 16; o > 0; o >>= 1) v += __shfl_xor(v, o, 32);
  int w = threadIdx.x >> 5;
  if ((threadIdx.x & 31) == 0) red[w] = v;
  __syncthreads();
  float s = red[0];
#pragma unroll
  for (int i = 1; i < 8; ++i) s += red[i];
  __syncthreads();
  return s;
}

// ---------------- 1) masked LayerNorm of x_a|x_b -> aN|bN -------------------
__global__ void ln_in_kernel(const float* __restrict__ xa, const float* __restrict__ xb,
                             const int* __restrict__ va, const int* __restrict__ vb,
                             const float* __restrict__ ga, const float* __restrict__ ba,
                             const float* __restrict__ gb, const float* __restrict__ bbv,
                             float* __restrict__ out) {
  __shared__ float red[8];
  int row = blockIdx.x;              // 0..8191
  int t = threadIdx.x;               // 0..255 == D index
  int half = row >> 12;              // 0 -> a, 1 -> b
  int r = row & 4095;
  const float* x = half ? xb : xa;
  const int* vv = half ? vb : va;
  const float* g = half ? gb : ga;
  const float* be = half ? bbv : ba;
  float v = x[r * Dn + t];
  float mu = block_sum_256(v, red) * (1.0f / Dn);
  float d = v - mu;
  float var = block_sum_256(d * d, red) * (1.0f / Dn);
  float y = d * rsqrtf(var + EPSv) * g[t] + be[t];
  out[row * Dn + t] = (vv[r] != 0) ? y : v;
}

// ---------------- 2) context: LN(cell @ Wctx + bctx) -> Cn ------------------
__global__ void ctx_kernel(const float* __restrict__ cell, const float* __restrict__ Wctx,
                           const float* __restrict__ bctx, const float* __restrict__ g,
                           const float* __restrict__ be, float* __restrict__ Cn) {
  __shared__ float cs[CINn];
  __shared__ float red[8];
  int row = blockIdx.x;              // 0..511 == b*16 + lc
  int b = row >> 4, lc = row & 15;
  int t = threadIdx.x;
  cs[t] = cell[b * CINn + t];
  cs[t + 256] = cell[b * CINn + 256 + t];
  __syncthreads();
  float acc = bctx[lc * Dn + t];
  const float* wp = Wctx + lc * Dn + t;
#pragma unroll 8
  for (int i = 0; i < CINn; ++i) acc += cs[i] * wp[i * (LCn * Dn)];
  float mu = block_sum_256(acc, red) * (1.0f / Dn);
  float d = acc - mu;
  float var = block_sum_256(d * d, red) * (1.0f / Dn);
  Cn[row * Dn + t] = d * rsqrtf(var + EPSv) * g[t] + be[t];
}

// ---------------- 3) GEMM: out[M,256] = A[M,256] @ W[256,256] + bias --------
// One wave computes one 16x16 output tile via V_WMMA_F32_16X16X4_F32.
// grid = (M/16, 2), block = 256 (8 waves); wave w -> ntile = by*8 + w.
__global__ void gemm256_wmma(const float* __restrict__ A, const float* __restrict__ W,
                             const float* __restrict__ bias, float* __restrict__ out) {
  __shared__ float As[16][257];      // padded: bank-conflict-free column reads
  int t = threadIdx.x;
  int mt = blockIdx.x;
#pragma unroll
  for (int i = 0; i < 16; ++i) As[i][t] = A[(mt * 16 + i) * Dn + t];
  __syncthreads();
  int lane = t & 31, w = t >> 5;
  int n0 = (blockIdx.y * 8 + w) * 16;
  int mR = lane & 15, hi = lane >> 4, kb = hi * 2;
  v8f acc = {};
  for (int kk = 0; kk < Dn; kk += 4) {
    v2f a, bf;
    a.x = As[mR][kk + kb];
    a.y = As[mR][kk + kb + 1];
    bf.x = W[(kk + kb) * Dn + n0 + mR];
    bf.y = W[(kk + kb + 1) * Dn + n0 + mR];
    acc = __builtin_amdgcn_wmma_f32_16x16x4_f32(false, a, false, bf, (short)0, acc,
                                                false, false);
  }
#pragma unroll
  for (int r = 0; r < 8; ++r) {
    int row = mt * 16 + r + 8 * hi;
    int col = n0 + mR;
    out[row * Dn + col] = acc[r] + bias[col];
  }
}

// ---------------- 4) tri-attention core -------------------------------------
// grid = (qt=8, h=8, dir*32+b = 64), block = 256 (8 waves).
// Per block: scores s[q,k,c] = ((K∘q)·C) with q pre-scaled, mask, online
// softmax over (k,c)=2048, then O[16,32] = (Σ_c attn)[16,128] @ V[128,32].
__global__ void tri_attn_kernel(const float* __restrict__ qab, const float* __restrict__ kab,
                                const float* __restrict__ vab, const float* __restrict__ cp,
                                const int* __restrict__ valid_a, const int* __restrict__ valid_b,
                                float* __restrict__ attnOut) {
  __shared__ float Qs[16][33];
  __shared__ float Ks[128][33];
  __shared__ float Vs[128][33];
  __shared__ float Cs[16][33];
  __shared__ float Amat[16][132];
  __shared__ float kexp[8][128];
  __shared__ float tMs[8][8];
  __shared__ int vqS[16];
  __shared__ int vkS[128];

  int qt = blockIdx.x;               // q tile (16 q's)
  int h = blockIdx.y;
  int dir = blockIdx.z >> 5;         // 0: out_a (q from a, k/v from b)
  int b = blockIdx.z & 31;
  int t = threadIdx.x;

  const float* Q = qab + (dir ? HALF_ELEMS : 0);
  const float* K = kab + (dir ? 0 : HALF_ELEMS);
  const float* V = vab + (dir ? 0 : HALF_ELEMS);
  const int* vQ = dir ? valid_b : valid_a;
  const int* vK = dir ? valid_a : valid_b;
  float* O = attnOut + (dir ? HALF_ELEMS : 0);

  // ---- stage tiles into LDS ----
#pragma unroll
  for (int i = 0; i < 16; ++i) {     // K, V: 128x32 each
    int j = i * 256 + t;
    int kr = j >> 5, kc = j & 31;
    Ks[kr][kc] = K[(b * Ln + kr) * Dn + h * DHn + kc];
    Vs[kr][kc] = V[(b * Ln + kr) * Dn + h * DHn + kc];
  }
#pragma unroll
  for (int i = 0; i < 2; ++i) {      // Q (scaled), C: 16x32 each
    int j = i * 256 + t;
    int qr = j >> 5, qc = j & 31;
    Qs[qr][qc] = Q[(b * Ln + qt * 16 + qr) * Dn + h * DHn + qc] * QSCALE;
    Cs[qr][qc] = cp[(b * LCn + qr) * Dn + h * DHn + qc];
  }
  if (t < 16) vqS[t] = vQ[b * Ln + qt * 16 + t];
  if (t < 128) vkS[t] = vK[b * Ln + t];
  __syncthreads();

  int lane = t & 31, w = t >> 5;
  int mR = lane & 15, hi = lane >> 4, kb = hi * 2;

  // ---- phase 1: scores + online softmax; each wave owns 2 q's ----
  for (int qq = 0; qq < 2; ++qq) {
    int qi = w * 2 + qq;
    int qvalid = vqS[qi];
    float m = -3.0e38f, s = 0.0f;
    for (int kt = 0; kt < 8; ++kt) {
      v8f acc = {};
#pragma unroll
      for (int kk = 0; kk < DHn; kk += 4) {
        v2f a, bf;
        a.x = Ks[kt * 16 + mR][kk + kb] * Qs[qi][kk + kb];
        a.y = Ks[kt * 16 + mR][kk + kb + 1] * Qs[qi][kk + kb + 1];
        bf.x = Cs[mR][kk + kb];      // B = C^T : B[kd][c] = Cs[c][kd], c = mR
        bf.y = Cs[mR][kk + kb + 1];
        acc = __builtin_amdgcn_wmma_f32_16x16x4_f32(false, a, false, bf, (short)0, acc,
                                                    false, false);
      }
      // mask + tile max
      float e[8];
      float tmax = -3.0e38f;
#pragma unroll
      for (int r = 0; r < 8; ++r) {
        int krow = kt * 16 + r + 8 * hi;
        float val = acc[r];
        if (!(qvalid && vkS[krow])) val = NEGv;
        e[r] = val;
        tmax = fmaxf(tmax, val);
      }
#pragma unroll
      for (int o = 16; o > 0; o >>= 1) tmax = fmaxf(tmax, __shfl_xor(tmax, o, 32));
      float mNew = fmaxf(m, tmax);
      float rescale = __expf(m - mNew);        // 0 on first tile
      float rs[8];
      float tsum = 0.0f;
#pragma unroll
      for (int r = 0; r < 8; ++r) {
        float ex = __expf(e[r] - mNew);
        ex += __shfl_xor(ex, 1, 16);           // row-sum over 16 c-lanes (per half)
        ex += __shfl_xor(ex, 2, 16);
        ex += __shfl_xor(ex, 4, 16);
        ex += __shfl_xor(ex, 8, 16);
        rs[r] = ex;
        tsum += ex;
      }
      tsum += __shfl_xor(tsum, 16, 32);        // combine halves
      s = s * rescale + tsum;
      m = mNew;
      if (mR == 0) {                           // lanes 0 and 16 record per-k sums
#pragma unroll
        for (int r = 0; r < 8; ++r) kexp[w][kt * 16 + r + 8 * hi] = rs[r];
        if (hi == 0) tMs[w][kt] = mNew;
      }
    }
    float invs = 1.0f / s;
#pragma unroll
    for (int i = 0; i < 4; ++i) {              // A[q,k] = (Σ_c attn[q,k,c])
      int k = i * 32 + lane;
      Amat[qi][k] = kexp[w][k] * __expf(tMs[w][k >> 4] - m) * invs;
    }
  }
  __syncthreads();

  // ---- phase 2: O[16,32] = Amat[16,128] @ V[128,32] (waves 0,1) ----
  if (w < 2) {
    int n0 = w * 16;
    v8f acc = {};
    for (int kk = 0; kk < Ln; kk += 4) {
      v2f a, bf;
      a.x = Amat[mR][kk + kb];
      a.y = Amat[mR][kk + kb + 1];
      bf.x = Vs[kk + kb][n0 + mR];
      bf.y = Vs[kk + kb + 1][n0 + mR];
      acc = __builtin_amdgcn_wmma_f32_16x16x4_f32(false, a, false, bf, (short)0, acc,
                                                  false, false);
    }
#pragma unroll
    for (int r = 0; r < 8; ++r) {
      int qrow = qt * 16 + r + 8 * hi;
      O[(b * Ln + qrow) * Dn + h * DHn + n0 + mR] = acc[r];
    }
  }
}

// ---------------- 5) residual + masked LN (in place on d_out) ---------------
__global__ void ln_out_kernel(const float* __restrict__ xa, const float* __restrict__ xb,
                              const int* __restrict__ va, const int* __restrict__ vb,
                              const float* __restrict__ ga, const float* __restrict__ ba,
                              const float* __restrict__ gb, const float* __restrict__ bbv,
                              float* __restrict__ out) {
  __shared__ float red[8];
  int row = blockIdx.x;
  int t = threadIdx.x;
  int half = row >> 12;
  int r = row & 4095;
  const float* x = half ? xb : xa;
  const int* vv = half ? vb : va;
  const float* g = half ? gb : ga;
  const float* be = half ? bbv : ba;
  float v = out[row * Dn + t] + x[r * Dn + t];   // residual
  float mu = block_sum_256(v, red) * (1.0f / Dn);
  float d = v - mu;
  float var = block_sum_256(d * d, red) * (1.0f / Dn);
  float y = d * rsqrtf(var + EPSv) * g[t] + be[t];
  out[row * Dn + t] = (vv[r] != 0) ? y : v;
}

// ---------------------------------------------------------------------------
extern "C" void kernel_launch(void* const* d_in, const int* in_sizes, int n_in,
                              void* d_out, int out_size, void* d_ws, size_t ws_size,
                              hipStream_t stream) {
  const float* x_a      = (const float*)d_in[0];
  const float* x_b      = (const float*)d_in[1];
  const int*   valid_a  = (const int*)d_in[2];
  const int*   valid_b  = (const int*)d_in[3];
  const float* cell     = (const float*)d_in[4];
  const float* ln_a_g   = (const float*)d_in[5];
  const float* ln_a_b   = (const float*)d_in[6];
  const float* ln_b_g   = (const float*)d_in[7];
  const float* ln_b_b   = (const float*)d_in[8];
  const float* ln_oa_g  = (const float*)d_in[9];
  const float* ln_oa_b  = (const float*)d_in[10];
  const float* ln_ob_g  = (const float*)d_in[11];
  const float* ln_ob_b  = (const float*)d_in[12];
  const float* ln_c_g   = (const float*)d_in[13];
  const float* ln_c_b   = (const float*)d_in[14];
  const float* Wctx     = (const float*)d_in[15];
  const float* bctx     = (const float*)d_in[16];
  const float* Wq       = (const float*)d_in[17];
  const float* bq       = (const float*)d_in[18];
  const float* Wk       = (const float*)d_in[19];
  const float* bk       = (const float*)d_in[20];
  const float* Wv       = (const float*)d_in[21];
  const float* bv       = (const float*)d_in[22];
  const float* Wc       = (const float*)d_in[23];
  const float* bc       = (const float*)d_in[24];
  const float* Wo       = (const float*)d_in[25];
  const float* bo       = (const float*)d_in[26];

  float* ws   = (float*)d_ws;
  float* abN  = ws;                               // [aN ; bN]     2*HALF
  float* qab  = ws + 2 * (size_t)HALF_ELEMS;      // [qa ; qb]     2*HALF
  float* kab  = ws + 4 * (size_t)HALF_ELEMS;      // [ka ; kb]     2*HALF
  float* vab  = ws + 6 * (size_t)HALF_ELEMS;      // [va ; vb]     2*HALF
  float* Cn   = ws + 8 * (size_t)HALF_ELEMS;      // 512*256
  float* cpj  = Cn + Bn * LCn * Dn;               // 512*256
  float* attn = cpj + Bn * LCn * Dn;              // [attnA ; attnB] 2*HALF
  float* outf = (float*)d_out;

  // 1) input masked LayerNorms
  ln_in_kernel<<<2 * Bn * Ln, 256, 0, stream>>>(x_a, x_b, valid_a, valid_b,
                                                ln_a_g, ln_a_b, ln_b_g, ln_b_b, abN);
  // 2) context tokenizer + LN
  ctx_kernel<<<Bn * LCn, 256, 0, stream>>>(cell, Wctx, bctx, ln_c_g, ln_c_b, Cn);
  // 3) projections (batched over [a;b], M = 8192)
  dim3 gBig(2 * Bn * Ln / 16, 2);
  gemm256_wmma<<<gBig, 256, 0, stream>>>(abN, Wq, bq, qab);
  gemm256_wmma<<<gBig, 256, 0, stream>>>(abN, Wk, bk, kab);
  gemm256_wmma<<<gBig, 256, 0, stream>>>(abN, Wv, bv, vab);
  dim3 gCtx(Bn * LCn / 16, 2);
  gemm256_wmma<<<gCtx, 256, 0, stream>>>(Cn, Wc, bc, cpj);
  // 4) tri-attention (both directions)
  dim3 gAttn(Ln / 16, Hn, 2 * Bn);
  tri_attn_kernel<<<gAttn, 256, 0, stream>>>(qab, kab, vab, cpj,
                                             valid_a, valid_b, attn);
  // 5) output projection (batched, writes d_out = [out_a ; out_b])
  gemm256_wmma<<<gBig, 256, 0, stream>>>(attn, Wo, bo, outf);
  // 6) residual + masked output LayerNorms (in place)
  ln_out_kernel<<<2 * Bn * Ln, 256, 0, stream>>>(x_a, x_b, valid_a, valid_b,
                                                 ln_oa_g, ln_oa_b, ln_ob_g, ln_ob_b, outf);
}